// ResidualVQ_4286377362151
// MI455X (gfx1250) — compile-verified
//
#include <hip/hip_runtime.h>
#include <hip/hip_bf16.h>

typedef __attribute__((ext_vector_type(2))) float v2f;
typedef __attribute__((ext_vector_type(8))) float v8f;

#define RVQ_B 8
#define RVQ_N 2048
#define RVQ_D 512
#define RVQ_Q 8
#define RVQ_C 1024
#define RVQ_M (RVQ_B * RVQ_N)          // 16384 rows
#define LDS_STRIDE 516                  // 512 + 4 pad -> conflict-free frag reads

// ---------------------------------------------------------------------------
// Kernel 1: squared norms of codewords  ee[c] = sum_d e[c,d]^2
// ---------------------------------------------------------------------------
__global__ __launch_bounds__(128) void vq_sqnorm(const float* __restrict__ E,
                                                 float* __restrict__ ee) {
    __shared__ float red[128];
    const int c = blockIdx.x;
    const int t = threadIdx.x;
    float4 v = ((const float4*)(E + (size_t)c * RVQ_D))[t];
    float s = v.x * v.x + v.y * v.y + v.z * v.z + v.w * v.w;
    red[t] = s;
    __syncthreads();
    for (int o = 64; o > 0; o >>= 1) {
        if (t < o) red[t] += red[t + o];
        __syncthreads();
    }
    if (t == 0) ee[c] = red[0];
}

// ---------------------------------------------------------------------------
// Kernel 2: fused GEMM (r . e^T via V_WMMA_F32_16X16X4_F32) + argmin over C.
// One wave (32 threads) per 16-row tile. A tile staged in LDS.
// Score = ||e||^2 - 2 r.e  (||r||^2 is constant per row, irrelevant to argmin)
// ---------------------------------------------------------------------------
__global__ __launch_bounds__(32) void vq_gemm_argmin(const float* __restrict__ R,
                                                     const float* __restrict__ E,
                                                     const float* __restrict__ ee,
                                                     int* __restrict__ idx) {
    __shared__ float lA[16 * LDS_STRIDE];

    const int lane = threadIdx.x;        // 0..31 (wave32)
    const int m0   = blockIdx.x * 16;

    // Stage A: 16 rows x 512 floats, coalesced float4 loads
    for (int i = lane; i < 16 * (RVQ_D / 4); i += 32) {
        const int row  = i >> 7;             // / (512/4)
        const int col4 = i & 127;
        float4 v = ((const float4*)(R + (size_t)(m0 + row) * RVQ_D))[col4];
        float* dst = &lA[row * LDS_STRIDE + col4 * 4];
        dst[0] = v.x; dst[1] = v.y; dst[2] = v.z; dst[3] = v.w;
    }
    __syncthreads();

    const int rsel  = lane & 15;             // A row / B column owned by this lane
    const int khalf = (lane >> 4) * 2;       // lanes 16..31 hold K+2,K+3

    float best[8];
    int   bestc[8];
#pragma unroll
    for (int v = 0; v < 8; ++v) { best[v] = 3.4e38f; bestc[v] = 0; }

    for (int c0 = 0; c0 < RVQ_C; c0 += 16) {
        v8f acc0 = {};
        v8f acc1 = {};
        const float* aptr = &lA[rsel * LDS_STRIDE + khalf];
        const float* bptr = E + (size_t)(c0 + rsel) * RVQ_D + khalf;

#pragma unroll 4
        for (int k0 = 0; k0 < RVQ_D; k0 += 8) {
            v2f a0 = *(const v2f*)(aptr);
            v2f b0 = *(const v2f*)(bptr);
            acc0 = __builtin_amdgcn_wmma_f32_16x16x4_f32(
                false, a0, false, b0, (short)0, acc0, false, false);
            v2f a1 = *(const v2f*)(aptr + 4);
            v2f b1 = *(const v2f*)(bptr + 4);
            acc1 = __builtin_amdgcn_wmma_f32_16x16x4_f32(
                false, a1, false, b1, (short)0, acc1, false, false);
            aptr += 8;
            bptr += 8;
        }

        const float eec  = ee[c0 + rsel];
        const int   ccol = c0 + rsel;
#pragma unroll
        for (int v = 0; v < 8; ++v) {
            // dist - ||r||^2 = ||e||^2 - 2 r.e ; strict < keeps earliest c on ties
            float s = fmaf(-2.0f, acc0[v] + acc1[v], eec);
            if (s < best[v]) { best[v] = s; bestc[v] = ccol; }
        }
    }

    // Cross-lane argmin within each 16-lane half (xor masks < 16 never cross bit 4).
    // Lanes 0-15 hold row m0+v, lanes 16-31 hold row m0+v+8 (C/D VGPR layout).
#pragma unroll
    for (int v = 0; v < 8; ++v) {
        float bs = best[v];
        int   bc = bestc[v];
        for (int sft = 1; sft < 16; sft <<= 1) {
            float os = __shfl_xor(bs, sft, 32);
            int   oc = __shfl_xor(bc, sft, 32);
            if (os < bs || (os == bs && oc < bc)) { bs = os; bc = oc; }
        }
        if (lane == 0)  idx[m0 + v]     = bc;
        if (lane == 16) idx[m0 + v + 8] = bc;
    }
}

// ---------------------------------------------------------------------------
// Kernel 3: gather codeword, accumulate quantized output, update residual,
// per-row commitment-loss partial, emit index as float. One block per row.
// ---------------------------------------------------------------------------
__global__ __launch_bounds__(128) void vq_update(const float* __restrict__ rin,
                                                 const float* __restrict__ E,
                                                 const int* __restrict__ idx,
                                                 float* __restrict__ resid,
                                                 float* __restrict__ qout,
                                                 float* __restrict__ idxout,
                                                 float* __restrict__ rowloss,
                                                 int qstage) {
    __shared__ float red[128];
    const int row = blockIdx.x;
    const int t   = threadIdx.x;
    const int c   = idx[row];

    const float4* ev   = (const float4*)(E + (size_t)c * RVQ_D);
    const float4* rv   = (const float4*)(rin + (size_t)row * RVQ_D);
    float4*       resv = (float4*)(resid + (size_t)row * RVQ_D);
    float4*       qv   = (float4*)(qout + (size_t)row * RVQ_D);

    float4 e = ev[t];
    float4 r = rv[t];
    float4 d;
    d.x = r.x - e.x; d.y = r.y - e.y; d.z = r.z - e.z; d.w = r.w - e.w;
    float err = d.x * d.x + d.y * d.y + d.z * d.z + d.w * d.w;

    resv[t] = d;                              // new residual
    if (qstage == 0) {
        qv[t] = e;                            // init accumulation
    } else {
        float4 qo = qv[t];
        qo.x += e.x; qo.y += e.y; qo.z += e.z; qo.w += e.w;
        qv[t] = qo;
    }

    red[t] = err;
    __syncthreads();
    for (int o = 64; o > 0; o >>= 1) {
        if (t < o) red[t] += red[t + o];
        __syncthreads();
    }
    if (t == 0) {
        rowloss[row] = red[0];
        idxout[(size_t)row * RVQ_Q + qstage] = (float)c;   // [B,N,Q] layout
    }
}

// ---------------------------------------------------------------------------
// Kernel 4: deterministic single-block reduction of row losses -> losses[q]
// ---------------------------------------------------------------------------
__global__ __launch_bounds__(256) void vq_loss(const float* __restrict__ rowloss,
                                               float* __restrict__ lossout,
                                               int n, float scale) {
    __shared__ float red[256];
    float s = 0.0f;
    for (int i = threadIdx.x; i < n; i += 256) s += rowloss[i];
    red[threadIdx.x] = s;
    __syncthreads();
    for (int o = 128; o > 0; o >>= 1) {
        if (threadIdx.x < o) red[threadIdx.x] += red[threadIdx.x + o];
        __syncthreads();
    }
    if (threadIdx.x == 0) *lossout = red[0] * scale;
}

// ---------------------------------------------------------------------------
extern "C" void kernel_launch(void* const* d_in, const int* in_sizes, int n_in,
                              void* d_out, int out_size, void* d_ws, size_t ws_size,
                              hipStream_t stream) {
    (void)in_sizes; (void)n_in; (void)out_size; (void)ws_size;

    const float* x  = (const float*)d_in[0];           // [B,N,D]
    const float* cb = (const float*)d_in[1];           // [Q,C,D]

    float* out_q    = (float*)d_out;                   // [B,N,D]
    float* out_idx  = out_q + (size_t)RVQ_M * RVQ_D;   // [B,N,Q] as float
    float* out_loss = out_idx + (size_t)RVQ_M * RVQ_Q; // [Q]

    float* resid   = (float*)d_ws;                     // [M,D]
    float* ee      = resid + (size_t)RVQ_M * RVQ_D;    // [C]
    int*   idx     = (int*)(ee + RVQ_C);               // [M]
    float* rowloss = (float*)(idx + RVQ_M);            // [M]

    const float scale = 1.0f / ((float)RVQ_M * (float)RVQ_D);

    for (int q = 0; q < RVQ_Q; ++q) {
        const float* Eq  = cb + (size_t)q * RVQ_C * RVQ_D;
        const float* rin = (q == 0) ? x : resid;

        vq_sqnorm<<<RVQ_C, 128, 0, stream>>>(Eq, ee);
        vq_gemm_argmin<<<RVQ_M / 16, 32, 0, stream>>>(rin, Eq, ee, idx);
        vq_update<<<RVQ_M, 128, 0, stream>>>(rin, Eq, idx, resid, out_q,
                                             out_idx, rowloss, q);
        vq_loss<<<1, 256, 0, stream>>>(rowloss, out_loss + q, RVQ_M, scale);
    }
}